// DistanceProbe_2886218023471
// MI455X (gfx1250) — compile-verified
//
#include <hip/hip_runtime.h>

// ---------------------------------------------------------------------------
// DistanceProbe on MI455X (gfx1250, wave32, WMMA).
//
// out[b,i,j] = max(||t_i||^2 + ||t_j||^2 - 2 t_i.t_j, 0),  t = batch @ proj
//
// All matmul work runs on V_WMMA_F32_16X16X4_F32 (f32 A/B/C) to preserve the
// reference's f32 semantics (the -2*dot term cancels catastrophically in
// lower precision). Compute-bound: 71 GFLOP f32 vs ~190MB of HBM traffic;
// the whole 50MB intermediate T lives in the 192MB L2.
//
// Wave tile is 64(M) x 64(N): 4x4 grid of 16x16 f32 accumulators (128 VGPRs).
// Per K=4 step: 8 float2 fragment loads feed 16 WMMAs -> ~16 FLOP/byte from
// L2, double the 32x64 variant, to keep the matrix pipes fed.
//
// Workspace layout (d_ws):
//   [0,                 50331648)  T     : [B*S, R] f32 projected points
//   [50331648,          50397184)  sq    : [B*S]    f32 row squared norms
//   [50397184,          52756480)  projT : [R, H]   f32 transposed projection
// ---------------------------------------------------------------------------

#define B_ 8
#define S_ 2048
#define H_ 768
#define R_ 768

typedef float v2f __attribute__((ext_vector_type(2)));
typedef float v8f __attribute__((ext_vector_type(8)));

// D = A(16x4 f32) * B(4x16 f32) + C(16x16 f32)
// Fragment layout (ISA 7.12.2):
//   A: lane l (l<16): M=l, holds K=0,1 in v[0],v[1]; lanes 16-31: K=2,3
//   B: mirrored: lane l&15 = N, half selects K pair -> contiguous f32x2 along K
//   C/D: VGPR r, lane l: element (M = r + 8*(l>>4), N = l&15)
static __device__ __forceinline__ v8f wmma_f32(v2f a, v2f b, v8f c) {
  return __builtin_amdgcn_wmma_f32_16x16x4_f32(
      /*neg_a=*/false, a, /*neg_b=*/false, b,
      /*c_mod=*/(short)0, c, /*reuse_a=*/false, /*reuse_b=*/false);
}

// ---------------------------------------------------------------------------
// Stage 0: projT[r][h] = proj[h][r]   (2.3 MB, LDS-tiled, bank-conflict free)
// ---------------------------------------------------------------------------
__global__ void __launch_bounds__(256)
transpose_proj_kernel(const float* __restrict__ proj, float* __restrict__ projT) {
  __shared__ float tile[32][33];
  const int bx = blockIdx.x * 32;  // R base
  const int by = blockIdx.y * 32;  // H base
  const int tx = threadIdx.x;      // 0..31
  const int ty = threadIdx.y;      // 0..7
#pragma unroll
  for (int i = 0; i < 32; i += 8)
    tile[ty + i][tx] = proj[(size_t)(by + ty + i) * R_ + (bx + tx)];
  __syncthreads();
#pragma unroll
  for (int i = 0; i < 32; i += 8)
    projT[(size_t)(bx + ty + i) * H_ + (by + tx)] = tile[tx][ty + i];
}

// ---------------------------------------------------------------------------
// Stage 1: T[m][n] = sum_k batch[m][k] * proj[k][n]   (M=B*S flat, N=R, K=H)
// Block = 4 waves (128 thr), block tile 256(M) x 64(N).
// Wave computes a 64x64 tile as a 4x4 grid of 16x16 WMMA accumulators.
// ---------------------------------------------------------------------------
__global__ void __launch_bounds__(128)
project_gemm_kernel(const float* __restrict__ batch,
                    const float* __restrict__ projT,
                    float* __restrict__ T) {
  const int lane = threadIdx.x & 31;
  const int wid  = threadIdx.x >> 5;
  const int half = lane >> 4;      // selects K pair {0,1} vs {2,3}
  const int lm   = lane & 15;      // M (for A) / N (for B,C)

  const int n0 = blockIdx.x * 64;
  const int m0 = blockIdx.y * 256 + wid * 64;

  v8f c[4][4];
#pragma unroll
  for (int t = 0; t < 4; ++t)
#pragma unroll
    for (int u = 0; u < 4; ++u)
#pragma unroll
      for (int e = 0; e < 8; ++e) c[t][u][e] = 0.0f;

  const float* aBase = batch + (size_t)(m0 + lm) * H_ + 2 * half;
  const float* bBase = projT + (size_t)(n0 + lm) * H_ + 2 * half;

#pragma unroll 2
  for (int k = 0; k < H_; k += 4) {
    v2f a[4], b[4];
#pragma unroll
    for (int t = 0; t < 4; ++t)
      a[t] = *(const v2f*)(aBase + (size_t)t * 16 * H_ + k);
#pragma unroll
    for (int u = 0; u < 4; ++u)
      b[u] = *(const v2f*)(bBase + (size_t)u * 16 * H_ + k);
#pragma unroll
    for (int t = 0; t < 4; ++t)
#pragma unroll
      for (int u = 0; u < 4; ++u)
        c[t][u] = wmma_f32(a[t], b[u], c[t][u]);
  }

#pragma unroll
  for (int t = 0; t < 4; ++t)
#pragma unroll
    for (int u = 0; u < 4; ++u)
#pragma unroll
      for (int r = 0; r < 8; ++r) {
        const int row = m0 + t * 16 + 8 * half + r;
        const int col = n0 + u * 16 + lm;
        T[(size_t)row * R_ + col] = c[t][u][r];
      }
}

// ---------------------------------------------------------------------------
// Stage 1.5: sq[row] = sum_r T[row][r]^2   (wave-per-row, wave32 shuffle tree)
// ---------------------------------------------------------------------------
__global__ void __launch_bounds__(256)
row_sq_kernel(const float* __restrict__ T, float* __restrict__ sq) {
  const int row  = blockIdx.x * 8 + (threadIdx.x >> 5);
  const int lane = threadIdx.x & 31;
  const float* p = T + (size_t)row * R_;
  float s = 0.0f;
#pragma unroll 4
  for (int r = lane; r < R_; r += 32) {
    const float v = p[r];
    s = fmaf(v, v, s);
  }
#pragma unroll
  for (int off = 16; off > 0; off >>= 1) s += __shfl_xor(s, off, 32);
  if (lane == 0) sq[row] = s;
}

// ---------------------------------------------------------------------------
// Stage 2: per batch b, Gram tile + fused distance epilogue.
// dots[i][j] = sum_r T[i][r]*T[j][r]  ->  B-frag B[k][j] = T[j][k] is a
// contiguous float2 along r, exactly like the A-frag. 64x64 wave tile.
// ---------------------------------------------------------------------------
__global__ void __launch_bounds__(128)
pair_dist_kernel(const float* __restrict__ T,
                 const float* __restrict__ sq,
                 float* __restrict__ out) {
  const int b    = blockIdx.z;
  const int lane = threadIdx.x & 31;
  const int wid  = threadIdx.x >> 5;
  const int half = lane >> 4;
  const int lm   = lane & 15;

  const int j0 = blockIdx.x * 64;
  const int i0 = blockIdx.y * 256 + wid * 64;

  const float* Tb  = T  + (size_t)b * S_ * R_;
  const float* sqb = sq + (size_t)b * S_;
  float* outb      = out + (size_t)b * S_ * S_;

  v8f c[4][4];
#pragma unroll
  for (int t = 0; t < 4; ++t)
#pragma unroll
    for (int u = 0; u < 4; ++u)
#pragma unroll
      for (int e = 0; e < 8; ++e) c[t][u][e] = 0.0f;

  const float* aBase = Tb + (size_t)(i0 + lm) * R_ + 2 * half;
  const float* bBase = Tb + (size_t)(j0 + lm) * R_ + 2 * half;

#pragma unroll 2
  for (int k = 0; k < R_; k += 4) {
    v2f a[4], bb[4];
#pragma unroll
    for (int t = 0; t < 4; ++t)
      a[t] = *(const v2f*)(aBase + (size_t)t * 16 * R_ + k);
#pragma unroll
    for (int u = 0; u < 4; ++u)
      bb[u] = *(const v2f*)(bBase + (size_t)u * 16 * R_ + k);
#pragma unroll
    for (int t = 0; t < 4; ++t)
#pragma unroll
      for (int u = 0; u < 4; ++u)
        c[t][u] = wmma_f32(a[t], bb[u], c[t][u]);
  }

  // Fused epilogue: d2 = sq_i + sq_j - 2*dot, clamped at 0.
  // Stores: lanes 0-15 hit consecutive j -> coalesced 64B per half-wave.
#pragma unroll
  for (int u = 0; u < 4; ++u) {
    const int j = j0 + u * 16 + lm;
    const float sj = sqb[j];
#pragma unroll
    for (int t = 0; t < 4; ++t) {
#pragma unroll
      for (int r = 0; r < 8; ++r) {
        const int i = i0 + t * 16 + 8 * half + r;
        const float d2 = sqb[i] + sj - 2.0f * c[t][u][r];
        outb[(size_t)i * S_ + j] = fmaxf(d2, 0.0f);
      }
    }
  }
}

// ---------------------------------------------------------------------------
extern "C" void kernel_launch(void* const* d_in, const int* in_sizes, int n_in,
                              void* d_out, int out_size, void* d_ws, size_t ws_size,
                              hipStream_t stream) {
  (void)in_sizes; (void)n_in; (void)out_size; (void)ws_size;

  const float* batch = (const float*)d_in[0];  // [B, S, H] f32
  const float* proj  = (const float*)d_in[1];  // [H, R]    f32
  float* out = (float*)d_out;                  // [B, S, S] f32

  char* ws = (char*)d_ws;
  float* T     = (float*)(ws);                                             // 50,331,648 B
  float* sqv   = (float*)(ws + (size_t)B_ * S_ * R_ * sizeof(float));      //     65,536 B
  float* projT = (float*)(ws + (size_t)B_ * S_ * R_ * sizeof(float)
                             + (size_t)B_ * S_ * sizeof(float));           //  2,359,296 B

  // Stage 0: transpose proj -> projT
  transpose_proj_kernel<<<dim3(R_ / 32, H_ / 32), dim3(32, 8), 0, stream>>>(proj, projT);

  // Stage 1: T = batch @ proj   (M = B*S flat, block tile 256x64)
  project_gemm_kernel<<<dim3(R_ / 64, (B_ * S_) / 256), 128, 0, stream>>>(batch, projT, T);

  // Stage 1.5: row squared norms
  row_sq_kernel<<<(B_ * S_) / 8, 256, 0, stream>>>(T, sqv);

  // Stage 2: Gram + fused distance epilogue (block tile 256x64 per batch)
  pair_dist_kernel<<<dim3(S_ / 64, S_ / 256, B_), 128, 0, stream>>>(T, sqv, out);
}